// Model_9139690406287
// MI455X (gfx1250) — compile-verified
//
#include <hip/hip_runtime.h>
#include <hip/hip_bf16.h>
#include <math.h>

typedef __attribute__((ext_vector_type(2))) float v2f;
typedef __attribute__((ext_vector_type(8))) float v8f;

#define N_NODES 1024
#define TOPK 8

// Native gfx1250 V_TANH_F32 when available (single trans op vs ~40-inst
// software tanh). tanh dominates the VALU budget here (~17M evals).
__device__ __forceinline__ float fast_tanh(float x) {
#if __has_builtin(__builtin_amdgcn_tanhf)
  return __builtin_amdgcn_tanhf(x);
#else
  return tanhf(x);
#endif
}

// Issue one async global->LDS 16-byte copy (ASYNCcnt-tracked).
__device__ __forceinline__ void async_copy_b128(void* lds_dst, const void* gsrc) {
  unsigned lds_addr = (unsigned)(unsigned long long)lds_dst;  // low 32 bits = LDS offset
  asm volatile("global_load_async_to_lds_b128 %0, %1, off"
               :: "v"(lds_addr), "v"(gsrc) : "memory");
}
__device__ __forceinline__ void async_wait0() {
  asm volatile("s_wait_asynccnt 0x0" ::: "memory");
}

// ---------------------------------------------------------------------------
// Kernel 1: node projections  n1 = tanh(3*(X@w1+b1)), n2 = tanh(3*(X@w2+b2))
// padded to k-stride 16 (zeros for k>=10) for the WMMA K=12 loop.
// ---------------------------------------------------------------------------
__global__ void proj_kernel(const float* __restrict__ x,
                            const float* __restrict__ w1, const float* __restrict__ b1,
                            const float* __restrict__ w2, const float* __restrict__ b2,
                            float* __restrict__ n1p, float* __restrict__ n2p, int z) {
  int idx = blockIdx.x * blockDim.x + threadIdx.x;   // over 1024*16
  if (idx >= N_NODES * 16) return;
  int n = idx >> 4, k = idx & 15;
  float o1 = 0.f, o2 = 0.f;
  if (k < 10) {
    const float* xr = x + (size_t)z * (N_NODES * 10) + n * 10;
    float s1 = b1[k], s2 = b2[k];
#pragma unroll
    for (int f = 0; f < 10; ++f) {
      float xv = xr[f];
      s1 += xv * w1[f * 10 + k];
      s2 += xv * w2[f * 10 + k];
    }
    o1 = fast_tanh(3.f * s1);
    o2 = fast_tanh(3.f * s2);
  }
  n1p[n * 16 + k] = o1;
  n2p[n * 16 + k] = o2;
}

// ---------------------------------------------------------------------------
// Kernel 2: adjacency tiles via V_WMMA_F32_16X16X4_F32.
// Block = 8 waves sharing row-tile tI; wave w handles col-tile tJ = jg*8+w.
// Panels staged to LDS with GLOBAL_LOAD_ASYNC_TO_LDS_B128 (panels are
// contiguous: padded row stride is 64B):
//   sA1/sA2 : n1/n2 rows of tile I   (1KB each)
//   sB1/sB2 : n1/n2 rows of J-group  (8KB each)
// Each wave: adj = relu(tanh(3*(S - S^T))), S = n1@n2^T; the S^T tile at
// (I,J) equals the (n2@n1^T) tile at (I,J) -> 6 WMMAs per tile, fused
// epilogue, only adj touches global memory.
// ---------------------------------------------------------------------------
__global__ void adj_wmma_kernel(const float* __restrict__ n1p,
                                const float* __restrict__ n2p,
                                float* __restrict__ adj) {
  __shared__ float sA1[16 * 16];
  __shared__ float sA2[16 * 16];
  __shared__ float sB1[128 * 16];
  __shared__ float sB2[128 * 16];

  int tI   = blockIdx.x >> 3;           // 0..63 row tile
  int jg   = blockIdx.x & 7;            // J group (8 col tiles)
  int wave = threadIdx.x >> 5;
  int lane = threadIdx.x & 31;
  int tid  = threadIdx.x;

  // Stage 4608 floats = 1152 x b128 chunks across 256 threads.
  for (int c = tid; c < 1152; c += 256) {
    const float* gsrc;
    float* ldst;
    if (c < 64)       { gsrc = n1p + tI * 256 + c * 4;          ldst = sA1 + c * 4; }
    else if (c < 128) { gsrc = n2p + tI * 256 + (c - 64) * 4;   ldst = sA2 + (c - 64) * 4; }
    else if (c < 640) { gsrc = n1p + jg * 2048 + (c - 128) * 4; ldst = sB1 + (c - 128) * 4; }
    else              { gsrc = n2p + jg * 2048 + (c - 640) * 4; ldst = sB2 + (c - 640) * 4; }
    async_copy_b128(ldst, gsrc);
  }
  async_wait0();
  __syncthreads();

  int r = lane & 15;
  int half = lane >> 4;
  int tJ = jg * 8 + wave;

  const float* a1 = sA1 + r * 16 + 2 * half;                 // S : A = n1_I
  const float* b2 = sB2 + (wave * 16 + r) * 16 + 2 * half;   // S : B = n2_J
  const float* a2 = sA2 + r * 16 + 2 * half;                 // S^T: A = n2_I
  const float* b1 = sB1 + (wave * 16 + r) * 16 + 2 * half;   // S^T: B = n1_J

  v8f c = {};
  v8f ct = {};
#pragma unroll
  for (int ks = 0; ks < 12; ks += 4) {
    v2f fa, fb, fa2, fb2;
    fa.x = a1[ks];  fa.y = a1[ks + 1];
    fb.x = b2[ks];  fb.y = b2[ks + 1];
    fa2.x = a2[ks]; fa2.y = a2[ks + 1];
    fb2.x = b1[ks]; fb2.y = b1[ks + 1];
    c  = __builtin_amdgcn_wmma_f32_16x16x4_f32(false, fa,  false, fb,  (short)0, c,  false, false);
    ct = __builtin_amdgcn_wmma_f32_16x16x4_f32(false, fa2, false, fb2, (short)0, ct, false, false);
  }

#pragma unroll
  for (int i = 0; i < 8; ++i) {
    float v = fast_tanh(3.f * (c[i] - ct[i]));
    v = v > 0.f ? v : 0.f;                 // relu
    int m = tI * 16 + i + 8 * half;        // D layout: M = vgpr + 8*(lane/16)
    int n = tJ * 16 + r;                   // N = lane%16
    adj[(size_t)m * N_NODES + n] = v;
  }
}

// ---------------------------------------------------------------------------
// Kernel 3: top-8 per row (stable, matches jax.lax.top_k tie-break = lowest
// index), plus symmetric-normalization degree d = rsqrt(1 + sum(top8 vals)).
// One 128-thread block per row; row staged to LDS via async copy.
// ---------------------------------------------------------------------------
__global__ void topk_kernel(const float* __restrict__ adj,
                            int* __restrict__ sp_idx, float* __restrict__ sp_val,
                            float* __restrict__ sp_d, int z) {
  __shared__ float sv[N_NODES];
  __shared__ float redv[128];
  __shared__ int   redi[128];
  int row = blockIdx.x;
  int t = threadIdx.x;
  const float* r = adj + (size_t)row * N_NODES;
  for (int c = t; c < N_NODES / 4; c += 128)     // 256 b128 chunks
    async_copy_b128(sv + c * 4, r + c * 4);
  async_wait0();
  __syncthreads();

  long base = ((long)z * N_NODES + row) * TOPK;
  for (int kk = 0; kk < TOPK; ++kk) {
    float bv = -1.f; int bi = N_NODES;     // values are >= 0
    for (int i = t; i < N_NODES; i += 128) {
      float v = sv[i];
      if (v > bv || (v == bv && i < bi)) { bv = v; bi = i; }
    }
    redv[t] = bv; redi[t] = bi;
    __syncthreads();
    for (int s = 64; s > 0; s >>= 1) {
      if (t < s) {
        float ov = redv[t + s]; int oi = redi[t + s];
        if (ov > redv[t] || (ov == redv[t] && oi < redi[t])) { redv[t] = ov; redi[t] = oi; }
      }
      __syncthreads();
    }
    if (t == 0) {
      sp_idx[base + kk] = redi[0];
      sp_val[base + kk] = redv[0];
      sv[redi[0]] = -1.f;                  // remove winner
    }
    __syncthreads();
  }
  if (t == 0) {
    float s = 1.f;                         // diagonal forced to 1
    for (int kk = 0; kk < TOPK; ++kk) s += sp_val[base + kk];
    sp_d[(long)z * N_NODES + row] = rsqrtf(s);   // s >= 1 so clip is a no-op
  }
}

// ---------------------------------------------------------------------------
// Kernel 4: feature transform  xw[z,n,:] = in[z,n,:] @ W   (tiny K, Fout)
// ---------------------------------------------------------------------------
__global__ void xw_kernel(const float* __restrict__ in, int zstride, int nstride,
                          const float* __restrict__ w, int K, int Fout,
                          float* __restrict__ xw, int total) {
  int idx = blockIdx.x * blockDim.x + threadIdx.x;
  if (idx >= total) return;
  int f = idx % Fout;
  int n = (idx / Fout) % N_NODES;
  int z = idx / (Fout * N_NODES);
  const float* row = in + (size_t)z * zstride + (size_t)n * nstride;
  float s = 0.f;
  for (int k = 0; k < K; ++k) s += row[k] * w[k * Fout + f];
  xw[((size_t)z * N_NODES + n) * Fout + f] = s;
}

// ---------------------------------------------------------------------------
// Kernel 5: sparse normalized aggregation
// out[n] = bias + d[n]*( d[n]*xw[n] + sum_j val_j * d[m_j] * xw[m_j] )
// relu_mode: 0=no relu, 1=relu, 2=relu only for stream 0 (z even; z = b*2+s)
// ---------------------------------------------------------------------------
__global__ void agg_kernel(const float* __restrict__ xw,
                           const int* __restrict__ sp_idx, const float* __restrict__ sp_val,
                           const float* __restrict__ sp_d, const float* __restrict__ bias,
                           int Fout, float* __restrict__ out, int relu_mode, int total) {
  int idx = blockIdx.x * blockDim.x + threadIdx.x;
  if (idx >= total) return;
  int f = idx % Fout;
  int n = (idx / Fout) % N_NODES;
  int z = idx / (Fout * N_NODES);
  const float* xwz = xw + (size_t)z * N_NODES * Fout;
  const float* dz = sp_d + (size_t)z * N_NODES;
  long base = ((long)z * N_NODES + n) * TOPK;
  float dn = dz[n];
  float s = dn * xwz[n * Fout + f];        // diagonal (value forced to 1)
#pragma unroll
  for (int j = 0; j < TOPK; ++j) {
    int m = sp_idx[base + j];
    float v = sp_val[base + j];            // 0 at idx==n, so no double-count
    s += v * dz[m] * xwz[m * Fout + f];
  }
  float o = dn * s + bias[f];
  bool relu = (relu_mode == 1) || (relu_mode == 2 && (z & 1) == 0);
  if (relu && o < 0.f) o = 0.f;
  out[((size_t)z * N_NODES + n) * Fout + f] = o;
}

// ---------------------------------------------------------------------------
// Kernel 6: layernorm over (N,8) per z; only row 0 survives to feat.
// ---------------------------------------------------------------------------
__global__ void ln_kernel(const float* __restrict__ h3,
                          const float* __restrict__ ln1_w, const float* __restrict__ ln1_b,
                          const float* __restrict__ ln2_w, const float* __restrict__ ln2_b,
                          float* __restrict__ feat) {
  __shared__ float ssum[256], ssq[256];
  int z = blockIdx.x, t = threadIdx.x;
  const float* h = h3 + (size_t)z * N_NODES * 8;
  float s = 0.f, sq = 0.f;
  for (int i = t; i < N_NODES * 8; i += 256) { float v = h[i]; s += v; sq += v * v; }
  ssum[t] = s; ssq[t] = sq;
  __syncthreads();
  for (int sft = 128; sft > 0; sft >>= 1) {
    if (t < sft) { ssum[t] += ssum[t + sft]; ssq[t] += ssq[t + sft]; }
    __syncthreads();
  }
  if (t < 8) {
    float m = ssum[0] * (1.f / 8192.f);
    float var = ssq[0] * (1.f / 8192.f) - m * m;
    float inv = rsqrtf(var + 1e-5f);
    int sId = z & 1, b = z >> 1;
    const float* lw = sId ? ln2_w : ln1_w;  // row 0 of (N,8)
    const float* lb = sId ? ln2_b : ln1_b;
    float o = (h[t] - m) * inv * lw[t] + lb[t];
    if (o < 0.f) o = 0.f;
    feat[b * 16 + sId * 8 + t] = o;
  }
}

// ---------------------------------------------------------------------------
// Kernel 7: head  out = (feat@lin1+b1)@lin2+b2   (8 batches, 2 outputs)
// ---------------------------------------------------------------------------
__global__ void head_kernel(const float* __restrict__ feat,
                            const float* __restrict__ l1w, const float* __restrict__ l1b,
                            const float* __restrict__ l2w, const float* __restrict__ l2b,
                            float* __restrict__ out) {
  int b = threadIdx.x;
  if (b >= 8) return;
  float t[8];
#pragma unroll
  for (int j = 0; j < 8; ++j) {
    float s = l1b[j];
#pragma unroll
    for (int i = 0; i < 16; ++i) s += feat[b * 16 + i] * l1w[i * 8 + j];
    t[j] = s;
  }
#pragma unroll
  for (int o = 0; o < 2; ++o) {
    float s = l2b[o];
#pragma unroll
    for (int j = 0; j < 8; ++j) s += t[j] * l2w[j * 2 + o];
    out[b * 2 + o] = s;
  }
}

extern "C" void kernel_launch(void* const* d_in, const int* in_sizes, int n_in,
                              void* d_out, int out_size, void* d_ws, size_t ws_size,
                              hipStream_t stream) {
  const float* x      = (const float*)d_in[0];
  const float* gc1_w1 = (const float*)d_in[1];
  const float* gc1_b1 = (const float*)d_in[2];
  const float* gc1_w2 = (const float*)d_in[3];
  const float* gc1_b2 = (const float*)d_in[4];
  const float* gc2_w1 = (const float*)d_in[5];
  const float* gc2_b1 = (const float*)d_in[6];
  const float* gc2_w2 = (const float*)d_in[7];
  const float* gc2_b2 = (const float*)d_in[8];
  const float* gcn1_w = (const float*)d_in[9];
  const float* gcn1_b = (const float*)d_in[10];
  const float* gcn2_w = (const float*)d_in[11];
  const float* gcn2_b = (const float*)d_in[12];
  const float* gcn3_w = (const float*)d_in[13];
  const float* gcn3_b = (const float*)d_in[14];
  const float* ln1_w  = (const float*)d_in[15];
  const float* ln1_b  = (const float*)d_in[16];
  const float* ln2_w  = (const float*)d_in[17];
  const float* ln2_b  = (const float*)d_in[18];
  const float* lin1_w = (const float*)d_in[19];
  const float* lin1_b = (const float*)d_in[20];
  const float* lin2_w = (const float*)d_in[21];
  const float* lin2_b = (const float*)d_in[22];
  float* out = (float*)d_out;

  // workspace carve-up (z = b*2 + s, 16 graphs)
  float* ws     = (float*)d_ws;
  float* n1p    = ws;                        // 1024*16
  float* n2p    = n1p + 16384;               // 1024*16
  float* adjf   = n2p + 16384;               // 1024*1024 (reused per z)
  int*   sp_idx = (int*)(adjf + 1024 * 1024);   // 16*1024*8
  float* sp_val = (float*)(sp_idx + 16 * 1024 * 8);
  float* sp_d   = sp_val + 16 * 1024 * 8;    // 16*1024
  float* xw     = sp_d + 16 * 1024;          // 16*1024*32 (reused per layer)
  float* h1     = xw + 16 * 1024 * 32;       // 16*1024*32
  float* h2     = h1 + 16 * 1024 * 32;       // 16*1024*16
  float* h3     = h2 + 16 * 1024 * 16;       // 16*1024*8
  float* feat   = h3 + 16 * 1024 * 8;        // 8*16

  // Phase 1: per-graph adjacency construction (sequential reuse of adjf)
  for (int z = 0; z < 16; ++z) {
    int s = z & 1;
    const float* w1 = s ? gc2_w1 : gc1_w1;
    const float* b1 = s ? gc2_b1 : gc1_b1;
    const float* w2 = s ? gc2_w2 : gc1_w2;
    const float* b2 = s ? gc2_b2 : gc1_b2;
    hipLaunchKernelGGL(proj_kernel, dim3(64), dim3(256), 0, stream,
                       x, w1, b1, w2, b2, n1p, n2p, z);
    hipLaunchKernelGGL(adj_wmma_kernel, dim3(512), dim3(256), 0, stream,
                       n1p, n2p, adjf);
    hipLaunchKernelGGL(topk_kernel, dim3(1024), dim3(128), 0, stream,
                       adjf, sp_idx, sp_val, sp_d, z);
  }

  // Phase 2: sparse GCN layers, batched over all 16 graphs
  // layer 1: 10 -> 32, relu
  {
    int total = 16 * 1024 * 32;
    hipLaunchKernelGGL(xw_kernel, dim3((total + 255) / 256), dim3(256), 0, stream,
                       x, 1024 * 10, 10, gcn1_w, 10, 32, xw, total);
    hipLaunchKernelGGL(agg_kernel, dim3((total + 255) / 256), dim3(256), 0, stream,
                       xw, sp_idx, sp_val, sp_d, gcn1_b, 32, h1, /*relu*/1, total);
  }
  // layer 2: 32 -> 16, relu only for stream 0
  {
    int total = 16 * 1024 * 16;
    hipLaunchKernelGGL(xw_kernel, dim3((total + 255) / 256), dim3(256), 0, stream,
                       h1, 1024 * 32, 32, gcn2_w, 32, 16, xw, total);
    hipLaunchKernelGGL(agg_kernel, dim3((total + 255) / 256), dim3(256), 0, stream,
                       xw, sp_idx, sp_val, sp_d, gcn2_b, 16, h2, /*relu*/2, total);
  }
  // layer 3: 16 -> 8, no relu (relu fused after LN)
  {
    int total = 16 * 1024 * 8;
    hipLaunchKernelGGL(xw_kernel, dim3((total + 255) / 256), dim3(256), 0, stream,
                       h2, 1024 * 16, 16, gcn3_w, 16, 8, xw, total);
    hipLaunchKernelGGL(agg_kernel, dim3((total + 255) / 256), dim3(256), 0, stream,
                       xw, sp_idx, sp_val, sp_d, gcn3_b, 8, h3, /*relu*/0, total);
  }

  // Phase 3: layernorm (row 0 only survives) + head
  hipLaunchKernelGGL(ln_kernel, dim3(16), dim3(256), 0, stream,
                     h3, ln1_w, ln1_b, ln2_w, ln2_b, feat);
  hipLaunchKernelGGL(head_kernel, dim3(1), dim3(32), 0, stream,
                     feat, lin1_w, lin1_b, lin2_w, lin2_b, out);
}